// SinkhornDistance_31353261261215
// MI455X (gfx1250) — compile-verified
//
#include <hip/hip_runtime.h>
#include <hip/hip_bf16.h>

typedef float v2f __attribute__((ext_vector_type(2)));
typedef float v8f __attribute__((ext_vector_type(8)));

#define BATCH   4
#define MSZ     1024
#define NSZ     1024
#define ITERS   40
#define EPS_INV 10.0f
#define INV_M   (1.0f / 1024.0f)   // uniform marginal a = b = 1/1024

#define KSEG    8                  // split-K segments per GEMV
#define SEGLEN  (MSZ / KSEG)       // 128 k-values per segment
#define NSTRIP  (BATCH * (MSZ / 16))        // 256 16-row/col stripes
#define NWAVE   (NSTRIP * KSEG)             // 2048 waves
#define NBLK    (NWAVE / 8)                 // 256 blocks x 8 waves

// ---------------------------------------------------------------------------
// 1) K = exp(-C/eps) into the P-region of d_out (overwritten with P at the
//    end); also resets the global-barrier counter for the persistent kernel.
// ---------------------------------------------------------------------------
__global__ void sink_setup(const float4* __restrict__ C4, float4* __restrict__ K4,
                           unsigned* __restrict__ barcnt) {
    int gid = blockIdx.x * blockDim.x + threadIdx.x;   // 0 .. 1M-1
    if (gid == 0) *barcnt = 0u;
    float4 c = C4[gid];
    float4 k;
    k.x = __expf(-EPS_INV * c.x);
    k.y = __expf(-EPS_INV * c.y);
    k.z = __expf(-EPS_INV * c.z);
    k.w = __expf(-EPS_INV * c.w);
    K4[gid] = k;
}

// Device-wide barrier: monotonic counter, target = phase * gridDim.x.
// Deterministic (results independent of arrival order); counter is reset by
// sink_setup before every persistent-kernel run.
__device__ __forceinline__ void gbar(unsigned* cnt, int phase) {
    __syncthreads();
    if (threadIdx.x == 0) {
        __threadfence();
        __hip_atomic_fetch_add(cnt, 1u, __ATOMIC_RELEASE, __HIP_MEMORY_SCOPE_AGENT);
        const unsigned target = (unsigned)phase * (unsigned)gridDim.x;
        while (__hip_atomic_load(cnt, __ATOMIC_ACQUIRE, __HIP_MEMORY_SCOPE_AGENT) < target)
            __builtin_amdgcn_s_sleep(1);
        __threadfence();
    }
    __syncthreads();
}

// ---------------------------------------------------------------------------
// 2) Persistent Sinkhorn loop: 40 x { u-phase, v-phase }, split-K over 8
//    segments, WMMA f32 16x16x4 reductions, fused partial-combine in the
//    LDS preload of the *next* phase.
//
//    Wave w (0..2047): strip = w & 255 -> (batch = strip>>6, tile16 = strip&63)
//                      seg   = w >> 8  -> k-range [seg*128, seg*128+128)
//    Every block (8 consecutive waves) shares one batch and one seg.
// ---------------------------------------------------------------------------
__global__ void sink_iter(const float* __restrict__ K,
                          float* __restrict__ Upart, float* __restrict__ Vpart,
                          float* __restrict__ U, float* __restrict__ V,
                          unsigned* __restrict__ barcnt) {
    __shared__ float sl[SEGLEN];

    const int tid   = threadIdx.x;
    const int lane  = tid & 31;                 // wave32
    const int w     = blockIdx.x * 8 + (tid >> 5);
    const int strip = w & (NSTRIP - 1);
    const int seg   = w >> 8;                   // uniform per block
    const int batch = strip >> 6;               // uniform per block
    const int t16   = (strip & 63) << 4;        // m0 (u-phase) / n0 (v-phase)
    const int hi    = lane >> 4;
    const int lo    = lane & 15;
    const int segbase = seg * SEGLEN;

    const float* Kb = K + (size_t)batch * MSZ * NSZ;

    // Seed partials so the combined v == 1:  sum_s Vpart[s] = INV_M.
    {
        int gid = blockIdx.x * 256 + tid;
        if (gid < KSEG * BATCH * NSZ)
            Vpart[gid] = (gid < BATCH * NSZ) ? INV_M : 0.0f;
    }
    int phase = 0;
    gbar(barcnt, ++phase);

    for (int it = 0; it < ITERS; ++it) {
        // ---- u-phase: Upart[seg][b][m] = sum_{j in seg} K[m][j] * v[j] ----
        for (int i = tid; i < SEGLEN; i += 256) {
            float s = 0.0f;
#pragma unroll
            for (int q = 0; q < KSEG; ++q)
                s += Vpart[q * (BATCH * NSZ) + batch * NSZ + segbase + i];
            sl[i] = INV_M / s;                  // fused combine + divide
        }
        __syncthreads();
        {
            const float* arow = Kb + (size_t)(t16 + lo) * NSZ + segbase + 2 * hi;
            v8f acc = {};
#pragma unroll 8
            for (int j0 = 0; j0 < SEGLEN; j0 += 4) {
                v2f a, b;
                float2 av = *(const float2*)(arow + j0);   // A: K 16x4 tile
                a.x = av.x;  a.y = av.y;
                b.x = sl[j0 + 2 * hi];                      // B: v replicated
                b.y = sl[j0 + 2 * hi + 1];
                acc = __builtin_amdgcn_wmma_f32_16x16x4_f32(
                          false, a, false, b, (short)0, acc, false, false);
            }
            float* Up = Upart + seg * (BATCH * MSZ) + batch * MSZ;
            if (lane == 0) {
#pragma unroll
                for (int g = 0; g < 8; ++g) Up[t16 + g] = acc[g];
            } else if (lane == 16) {
#pragma unroll
                for (int g = 0; g < 8; ++g) Up[t16 + 8 + g] = acc[g];
            }
        }
        gbar(barcnt, ++phase);

        // ---- v-phase: Vpart[seg][b][n] = sum_{m in seg} K[m][n] * u[m] ----
        for (int i = tid; i < SEGLEN; i += 256) {
            float s = 0.0f;
#pragma unroll
            for (int q = 0; q < KSEG; ++q)
                s += Upart[q * (BATCH * MSZ) + batch * MSZ + segbase + i];
            sl[i] = INV_M / s;
        }
        __syncthreads();
        {
            const float* bcol = Kb + (size_t)(segbase + 2 * hi) * NSZ + t16 + lo;
            v8f acc = {};
#pragma unroll 8
            for (int k0 = 0; k0 < SEGLEN; k0 += 4) {
                v2f a, b;
                a.x = sl[k0 + 2 * hi];                      // A: u replicated
                a.y = sl[k0 + 2 * hi + 1];
                b.x = bcol[(size_t)k0 * NSZ];               // B: K 4x16 tile
                b.y = bcol[(size_t)k0 * NSZ + NSZ];
                acc = __builtin_amdgcn_wmma_f32_16x16x4_f32(
                          false, a, false, b, (short)0, acc, false, false);
            }
            float* Vp = Vpart + seg * (BATCH * NSZ) + batch * NSZ;
            if (lane < 16) Vp[t16 + lane] = acc[0];
        }
        gbar(barcnt, ++phase);
    }

    // Materialize final scaling vectors u, v for the P/distance pass.
    {
        int gid = blockIdx.x * 256 + tid;
        if (gid < BATCH * MSZ) {
            float s = 0.0f;
#pragma unroll
            for (int q = 0; q < KSEG; ++q) s += Upart[q * (BATCH * MSZ) + gid];
            U[gid] = INV_M / s;
        } else if (gid < 2 * BATCH * MSZ) {
            int g2 = gid - BATCH * MSZ;
            float s = 0.0f;
#pragma unroll
            for (int q = 0; q < KSEG; ++q) s += Vpart[q * (BATCH * NSZ) + g2];
            V[g2] = INV_M / s;
        }
    }
}

// ---------------------------------------------------------------------------
// 3) P = u_m * K * v_n (in place over K = d_out P-region) + per-block
//    partial of sum(P*C) for a deterministic reduction.
// ---------------------------------------------------------------------------
__global__ void sink_final(const float4* __restrict__ C4, const float* __restrict__ U,
                           const float* __restrict__ V, float4* __restrict__ P4,
                           float* __restrict__ partial) {
    __shared__ float red[256];
    const int gid  = blockIdx.x * 256 + threadIdx.x;   // float4 index
    const int flat = gid << 2;
    const int batch = flat >> 20;
    const int rem   = flat & ((1 << 20) - 1);
    const int m     = rem >> 10;
    const int n     = rem & 1023;

    const float  u  = U[batch * MSZ + m];
    const float4 v4 = *(const float4*)(V + batch * NSZ + n);
    const float4 k  = P4[gid];          // K stored here by sink_setup
    const float4 c  = C4[gid];

    float4 p;
    p.x = u * k.x * v4.x;
    p.y = u * k.y * v4.y;
    p.z = u * k.z * v4.z;
    p.w = u * k.w * v4.w;
    P4[gid] = p;                        // overwrite K with P (output)

    red[threadIdx.x] = p.x * c.x + p.y * c.y + p.z * c.z + p.w * c.w;
    __syncthreads();
    for (int s = 128; s > 0; s >>= 1) {
        if (threadIdx.x < s) red[threadIdx.x] += red[threadIdx.x + s];
        __syncthreads();
    }
    if (threadIdx.x == 0) partial[blockIdx.x] = red[0];
}

__global__ void sink_reduce(const float* __restrict__ partial, float* __restrict__ dist) {
    __shared__ float red[256];
    const int b = blockIdx.x;
    float acc = 0.0f;
    for (int i = threadIdx.x; i < 1024; i += 256) acc += partial[b * 1024 + i];
    red[threadIdx.x] = acc;
    __syncthreads();
    for (int s = 128; s > 0; s >>= 1) {
        if (threadIdx.x < s) red[threadIdx.x] += red[threadIdx.x + s];
        __syncthreads();
    }
    if (threadIdx.x == 0) dist[b] = red[0];
}

// ---------------------------------------------------------------------------
extern "C" void kernel_launch(void* const* d_in, const int* in_sizes, int n_in,
                              void* d_out, int out_size, void* d_ws, size_t ws_size,
                              hipStream_t stream) {
    (void)in_sizes; (void)n_in; (void)out_size; (void)ws_size;

    const float* C = (const float*)d_in[0];
    float* out     = (float*)d_out;
    float* Kbuf    = out + 4;                       // 4M floats: K, then P

    float* Upart   = (float*)d_ws;                  // KSEG * 4096
    float* Vpart   = Upart + KSEG * BATCH * MSZ;    // KSEG * 4096
    float* U       = Vpart + KSEG * BATCH * NSZ;    // 4096
    float* V       = U + BATCH * MSZ;               // 4096
    float* partial = V + BATCH * NSZ;               // 4096 block partials
    unsigned* barcnt = (unsigned*)(partial + 4096); // barrier counter

    sink_setup<<<4096, 256, 0, stream>>>((const float4*)C, (float4*)Kbuf, barcnt);
    sink_iter<<<NBLK, 256, 0, stream>>>(Kbuf, Upart, Vpart, U, V, barcnt);
    sink_final<<<4096, 256, 0, stream>>>((const float4*)C, U, V,
                                         (float4*)Kbuf, partial);
    sink_reduce<<<BATCH, 256, 0, stream>>>(partial, out);
}